// RoundRobinGate_80496277062245
// MI455X (gfx1250) — compile-verified
//
#include <hip/hip_runtime.h>
#include <hip/hip_bf16.h>

// RoundRobinGate: output depends only on shapes. d_out (float) layout:
//   [0]                    : scalar loss = 0.0
//   [1, 1+2^27)            : one-hot float tensor (g=4, s=4096, E=16, C=512)
//   [1+2^27, 1+2^28)       : bool mask as 0.0/1.0 (same pattern)
// Total = 2^28 + 1 floats ~= 1.074 GB of streaming stores; 32768 ones total.
// Pure store-bandwidth problem: floor ~= 1.074e9 B / 23.3 TB/s ~= 46 us.
// WMMA is irrelevant (0 useful FLOPs); the CDNA5 path that matters is the
// async LDS->global B128 store stream (ASYNCcnt-pipelined, no VGPR data).

typedef float v4f __attribute__((ext_vector_type(4)));
// Exact param type of the async-store builtin (from hipcc diagnostic):
//   '__attribute__((__vector_size__(4 * sizeof(int)))) int *'
typedef int v4i __attribute__((vector_size(4 * sizeof(int))));

#define AS_LDS __attribute__((address_space(3)))

#ifndef __has_builtin
#define __has_builtin(x) 0
#endif

#if __has_builtin(__builtin_amdgcn_global_store_async_from_lds_b128)
#define RR_USE_ASYNC_LDS 1
#else
#define RR_USE_ASYNC_LDS 0
#endif

// Zero-fill [0, 2^28) floats with 16B stores. Covers d_out[0] (= 0.0).
// Launch: 4096 blocks x 256 threads. Each block-iteration covers 1024 chunks
// (16 KB); 65536 groups total -> 16 uniform iterations per block.
__global__ void rr_zero_fill(float* __restrict__ out) {
    const long long groups  = 65536;            // 2^26 chunks / 1024
    const long long gstride = gridDim.x;

#if RR_USE_ASYNC_LDS
    // CDNA5 async path: stage 512B of zeros in LDS once, then blast
    // global_store_async_from_lds_b128 (no VGPR data, up to 63 in flight).
    __shared__ __align__(16) float zbuf[128];   // 512 B
    if (threadIdx.x < 128) zbuf[threadIdx.x] = 0.0f;
    __syncthreads();
    AS_LDS v4i* lsrc = (AS_LDS v4i*)(&zbuf[(threadIdx.x & 31) * 4]);

    for (long long grp = blockIdx.x; grp < groups; grp += gstride) {
        // grp*1024 chunks * 4 floats = grp<<12 floats; thread t at chunk t.
        v4i* gdst = (v4i*)(out + (grp << 12) + ((long long)threadIdx.x << 2));
        // Thread t covers chunks t, t+256, t+512, t+768 via imm offsets:
        // every instruction is 4 KB fully-coalesced across the block.
        __builtin_amdgcn_global_store_async_from_lds_b128(gdst, lsrc, 0, 0);
        __builtin_amdgcn_global_store_async_from_lds_b128(gdst, lsrc, 4096, 0);
        __builtin_amdgcn_global_store_async_from_lds_b128(gdst, lsrc, 8192, 0);
        __builtin_amdgcn_global_store_async_from_lds_b128(gdst, lsrc, 12288, 0);
    }
#if __has_builtin(__builtin_amdgcn_s_wait_asynccnt)
    __builtin_amdgcn_s_wait_asynccnt(0);
#else
    asm volatile("s_wait_asynccnt 0" ::: "memory");
#endif
#else
    // Fallback: 16B non-temporal stores (global_store_b128 th:TH_STORE_NT) —
    // output is 1 GB >> 192 MB L2, so stream past the cache.
    const v4f z = {0.0f, 0.0f, 0.0f, 0.0f};
    for (long long grp = blockIdx.x; grp < groups; grp += gstride) {
        float* base = out + (grp << 12) + ((long long)threadIdx.x << 2);
        __builtin_nontemporal_store(z, (v4f*)(base));
        __builtin_nontemporal_store(z, (v4f*)(base + 1024));
        __builtin_nontemporal_store(z, (v4f*)(base + 2048));
        __builtin_nontemporal_store(z, (v4f*)(base + 3072));
    }
#endif
}

// Write the 32768 ones (16384 per output region) + the single tail element.
// Flat offset within a region: p = g*2^25 + tok*2^13 + (tok%16)*2^9 + tok/16.
__global__ void rr_write_ones(float* __restrict__ out) {
    const int t = blockIdx.x * blockDim.x + threadIdx.x; // [0, 32768)
    const int region = t >> 14;        // 0 = float one-hot, 1 = bool mask
    const int r      = t & 16383;
    const int g      = r >> 12;        // [0, 4)
    const int tok    = r & 4095;       // [0, 4096)
    const long long p = ((long long)g << 25) | ((long long)tok << 13) |
                        ((long long)(tok & 15) << 9) | (long long)(tok >> 4);
    const long long idx = 1LL + ((long long)region << 27) + p;
    out[idx] = 1.0f;
    if (t == 0) {
        // Last element (index 2^28) not covered by the 16B-chunk fill; its
        // value is 0.0 (tok=4095 -> slot 255 != c 511).
        out[268435456LL] = 0.0f;
    }
}

extern "C" void kernel_launch(void* const* d_in, const int* in_sizes, int n_in,
                              void* d_out, int out_size, void* d_ws, size_t ws_size,
                              hipStream_t stream) {
    (void)d_in; (void)in_sizes; (void)n_in; (void)d_ws; (void)ws_size; (void)out_size;
    float* out = (float*)d_out;

    // 4096 blocks x 256 threads (8 waves/block): each block streams 16
    // groups x 16 KB = 256 KB; plenty of waves to saturate 23.3 TB/s HBM.
    rr_zero_fill<<<4096, 256, 0, stream>>>(out);

    // 32768 one-writes, one thread each.
    rr_write_ones<<<128, 256, 0, stream>>>(out);
}